// MultiHeadGATLayer_78546361909459
// MI455X (gfx1250) — compile-verified
//
#include <hip/hip_runtime.h>
#include <hip/hip_bf16.h>

// ---------------- problem constants ----------------
#define NNODE  50000
#define NEDGE  800000
#define NTOT   850000      // edges + self loops
#define KDIM   256         // IN_C
#define HC     128         // HEADS * OUT_C
#define NEG_SLOPE 0.2f
#define LN_EPS 1e-5f

typedef float v2f __attribute__((ext_vector_type(2)));
typedef float v8f __attribute__((ext_vector_type(8)));

#define LDA 260            // padded LDS row stride (floats) to avoid bank conflicts

// ---------------------------------------------------------------------------
// Zero-fill helper (accumulators must start at 0 every call).
// ---------------------------------------------------------------------------
__global__ void gat_zero_kernel(float* __restrict__ p, int n) {
    int i = blockIdx.x * blockDim.x + threadIdx.x;
    if (i < n) p[i] = 0.0f;
}

// ---------------------------------------------------------------------------
// Fused dual GEMM: h = x @ W ; idn = x @ W_res      (M=50000, K=256, N=128)
// Block = 256 threads = 8 waves. Block computes 16 rows x 128 cols for BOTH
// outputs. A-tile (16x256 of x) staged once in LDS and shared by all waves;
// each wave owns a 16-column strip and runs V_WMMA_F32_16X16X4_F32 (full
// fp32 precision; the layer is HBM-bound so the f32 matrix path is optimal).
// ---------------------------------------------------------------------------
__global__ __launch_bounds__(256) void gat_gemm_dual_kernel(
    const float* __restrict__ x,
    const float* __restrict__ W,
    const float* __restrict__ Wr,
    float* __restrict__ h,
    float* __restrict__ idn)
{
    __shared__ float As[16 * LDA];

    const int tid  = threadIdx.x;
    const int row0 = blockIdx.x * 16;

    // cooperative load of 16x256 A tile (1024 float4, 4 per thread), coalesced
    {
        const float4* xv = (const float4*)(x + (size_t)row0 * KDIM);
        #pragma unroll
        for (int i = 0; i < 4; ++i) {
            int flat = tid + i * 256;       // float4 index within tile
            int r    = flat >> 6;           // row (64 float4 per row)
            int c4   = flat & 63;
            ((float4*)(As + r * LDA))[c4] = xv[flat];
        }
    }
    __syncthreads();

    const int wave = tid >> 5;        // 0..7 -> 16-column strip
    const int lane = tid & 31;
    const int g    = lane >> 4;       // lane group: K offset of 2*g
    const int m    = lane & 15;       // A row within tile / B column
    const int col0 = wave * 16;

    v8f accW = {};
    v8f accR = {};

    const float* arow = As + m * LDA + 2 * g;   // A: VGPR v holds K = k0 + 2g + v

    #pragma unroll 4
    for (int k0 = 0; k0 < KDIM; k0 += 4) {
        v2f a = *(const v2f*)(arow + k0);       // 8B-aligned ds_load

        const int kk = k0 + 2 * g;              // B: VGPR v holds K = k0 + 2g + v
        v2f bW, bR;
        bW.x = W [(size_t)kk       * HC + col0 + m];
        bW.y = W [(size_t)(kk + 1) * HC + col0 + m];
        bR.x = Wr[(size_t)kk       * HC + col0 + m];
        bR.y = Wr[(size_t)(kk + 1) * HC + col0 + m];

        accW = __builtin_amdgcn_wmma_f32_16x16x4_f32(
                   false, a, false, bW, (short)0, accW, false, false);
        accR = __builtin_amdgcn_wmma_f32_16x16x4_f32(
                   false, a, false, bR, (short)0, accR, false, false);
    }

    // C/D layout: VGPR r -> row (r + 8*laneGroup), lane[3:0] -> column
    #pragma unroll
    for (int r = 0; r < 8; ++r) {
        const int row = row0 + r + 8 * g;
        h  [(size_t)row * HC + col0 + m] = accW[r];
        idn[(size_t)row * HC + col0 + m] = accR[r];
    }
}

// ---------------------------------------------------------------------------
// Attention coefficients: a_src[n,h] = <h[n,h,:], att_src[h,:]> (and dst).
// One wave per node, lane l owns channel l of every head; wave32 shfl_xor
// reductions produce the 4-per-head dot products.
// ---------------------------------------------------------------------------
__global__ __launch_bounds__(256) void gat_attn_coef_kernel(
    const float* __restrict__ h,
    const float* __restrict__ att_src,
    const float* __restrict__ att_dst,
    float* __restrict__ a_src,
    float* __restrict__ a_dst)
{
    const int node = blockIdx.x * 8 + (threadIdx.x >> 5);
    if (node >= NNODE) return;
    const int lane = threadIdx.x & 31;

    const float* hr = h + (size_t)node * HC;
    float ps[4], pd[4];
    #pragma unroll
    for (int hd = 0; hd < 4; ++hd) {
        float v = hr[hd * 32 + lane];
        ps[hd] = v * att_src[hd * 32 + lane];
        pd[hd] = v * att_dst[hd * 32 + lane];
    }
    #pragma unroll
    for (int off = 16; off > 0; off >>= 1) {
        #pragma unroll
        for (int hd = 0; hd < 4; ++hd) {
            ps[hd] += __shfl_xor(ps[hd], off, 32);
            pd[hd] += __shfl_xor(pd[hd], off, 32);
        }
    }
    if (lane == 0) {
        #pragma unroll
        for (int hd = 0; hd < 4; ++hd) {
            a_src[node * 4 + hd] = ps[hd];
            a_dst[node * 4 + hd] = pd[hd];
        }
    }
}

// ---------------------------------------------------------------------------
// Single edge pass (incl. implicit self-loops for e >= NEDGE):
//   w_h = exp(leaky_relu(a_src[s,h] + a_dst[d,h]))
//   acc[d, h*32+c] += w_h * hfeat[s, h*32+c]   (global_atomic_add_f32)
//   denom[d, h]    += w_h
// Softmax normalization is deferred to the finalize pass (alpha = w/denom).
// One wave per edge; lane l handles channel l of each head.
// ---------------------------------------------------------------------------
__global__ __launch_bounds__(256) void gat_edge_scatter_kernel(
    const int* __restrict__ ei,          // [2, NEDGE] flat: src row then dst row
    const float* __restrict__ hfeat,
    const float* __restrict__ a_src,
    const float* __restrict__ a_dst,
    float* __restrict__ acc,             // d_out used as accumulator
    float* __restrict__ denom)
{
    const int e = blockIdx.x * 8 + (threadIdx.x >> 5);
    if (e >= NTOT) return;
    const int lane = threadIdx.x & 31;

    int s, d;
    if (e < NEDGE) { s = ei[e]; d = ei[NEDGE + e]; }
    else           { s = d = e - NEDGE; }

    const float4 as4 = *(const float4*)(a_src + (size_t)s * 4);
    const float4 ad4 = *(const float4*)(a_dst + (size_t)d * 4);

    float lg[4] = { as4.x + ad4.x, as4.y + ad4.y, as4.z + ad4.z, as4.w + ad4.w };
    float w[4];
    #pragma unroll
    for (int hd = 0; hd < 4; ++hd) {
        float v = lg[hd];
        v = (v >= 0.0f) ? v : NEG_SLOPE * v;   // leaky_relu
        w[hd] = __expf(v);                     // max-shift omitted (exact softmax identity)
    }

    const float* hs = hfeat + (size_t)s * HC;
    float*       od = acc   + (size_t)d * HC;
    #pragma unroll
    for (int hd = 0; hd < 4; ++hd) {
        float v = hs[hd * 32 + lane];
        atomicAdd(od + hd * 32 + lane, w[hd] * v);
    }
    if (lane < 4) {
        float wl = (lane == 0) ? w[0] : (lane == 1) ? w[1] : (lane == 2) ? w[2] : w[3];
        atomicAdd(denom + (size_t)d * 4 + lane, wl);
    }
}

// ---------------------------------------------------------------------------
// Finalize (in-place on d_out): y = acc/denom + bias + idn + b_res,
// LayerNorm(128) with gamma/beta, ReLU. One wave per node, 4 channels/lane,
// two-pass mean/variance via wave reductions.
// ---------------------------------------------------------------------------
__global__ __launch_bounds__(256) void gat_finalize_kernel(
    float* __restrict__ out,
    const float* __restrict__ idn,
    const float* __restrict__ denom,
    const float* __restrict__ bias,
    const float* __restrict__ b_res,
    const float* __restrict__ gamma,
    const float* __restrict__ beta)
{
    const int node = blockIdx.x * 8 + (threadIdx.x >> 5);
    if (node >= NNODE) return;
    const int lane = threadIdx.x & 31;
    const int c0   = lane * 4;                    // channels c0..c0+3 (same head)

    float*       orow = out + (size_t)node * HC;
    const float* irow = idn + (size_t)node * HC;

    const float4 a4 = *(const float4*)(orow + c0);
    const float4 i4 = *(const float4*)(irow + c0);
    const float4 b4 = *(const float4*)(bias + c0);
    const float4 r4 = *(const float4*)(b_res + c0);

    const float inv = 1.0f / denom[(size_t)node * 4 + (lane >> 3)];

    float y[4];
    y[0] = a4.x * inv + i4.x + b4.x + r4.x;
    y[1] = a4.y * inv + i4.y + b4.y + r4.y;
    y[2] = a4.z * inv + i4.z + b4.z + r4.z;
    y[3] = a4.w * inv + i4.w + b4.w + r4.w;

    // pass 1: mean
    float s = y[0] + y[1] + y[2] + y[3];
    #pragma unroll
    for (int off = 16; off > 0; off >>= 1) s += __shfl_xor(s, off, 32);
    const float mu = s * (1.0f / 128.0f);

    // pass 2: variance of (y - mu)
    float ss = 0.0f;
    #pragma unroll
    for (int k = 0; k < 4; ++k) { float t = y[k] - mu; ss += t * t; }
    #pragma unroll
    for (int off = 16; off > 0; off >>= 1) ss += __shfl_xor(ss, off, 32);
    const float rstd = rsqrtf(ss * (1.0f / 128.0f) + LN_EPS);

    const float4 g4 = *(const float4*)(gamma + c0);
    const float4 t4 = *(const float4*)(beta + c0);

    float4 o;
    o.x = fmaxf((y[0] - mu) * rstd * g4.x + t4.x, 0.0f);
    o.y = fmaxf((y[1] - mu) * rstd * g4.y + t4.y, 0.0f);
    o.z = fmaxf((y[2] - mu) * rstd * g4.z + t4.z, 0.0f);
    o.w = fmaxf((y[3] - mu) * rstd * g4.w + t4.w, 0.0f);
    *(float4*)(orow + c0) = o;
}

// ---------------------------------------------------------------------------
extern "C" void kernel_launch(void* const* d_in, const int* in_sizes, int n_in,
                              void* d_out, int out_size, void* d_ws, size_t ws_size,
                              hipStream_t stream) {
    const float* x       = (const float*)d_in[0];
    const int*   ei      = (const int*)  d_in[1];   // int32 (JAX x64 disabled)
    const float* W       = (const float*)d_in[2];
    const float* att_src = (const float*)d_in[3];
    const float* att_dst = (const float*)d_in[4];
    const float* bias    = (const float*)d_in[5];
    const float* W_res   = (const float*)d_in[6];
    const float* b_res   = (const float*)d_in[7];
    const float* gamma   = (const float*)d_in[8];
    const float* beta    = (const float*)d_in[9];

    float* out = (float*)d_out;

    // workspace layout (floats)
    float* ws     = (float*)d_ws;
    float* h      = ws;                       // 50000*128 = 6,400,000
    float* idn    = ws + 6400000;             // 6,400,000
    float* a_src  = ws + 12800000;            // 200,000
    float* a_dst  = ws + 13000000;            // 200,000
    float* denom  = ws + 13200000;            // 200,000   (total 53.6 MB)

    const int NOUT = NNODE * HC;              // 6,400,000
    gat_zero_kernel<<<(NOUT + 255) / 256, 256, 0, stream>>>(out, NOUT);
    gat_zero_kernel<<<(NNODE * 4 + 255) / 256, 256, 0, stream>>>(denom, NNODE * 4);

    gat_gemm_dual_kernel<<<NNODE / 16, 256, 0, stream>>>(x, W, W_res, h, idn);

    gat_attn_coef_kernel<<<(NNODE + 7) / 8, 256, 0, stream>>>(h, att_src, att_dst,
                                                              a_src, a_dst);

    gat_edge_scatter_kernel<<<(NTOT + 7) / 8, 256, 0, stream>>>(ei, h, a_src, a_dst,
                                                                out, denom);

    gat_finalize_kernel<<<(NNODE + 7) / 8, 256, 0, stream>>>(out, idn, denom, bias,
                                                             b_res, gamma, beta);
}